// GraphConvolution_4595615007148
// MI455X (gfx1250) — compile-verified
//
#include <hip/hip_runtime.h>

typedef __attribute__((ext_vector_type(2))) float v2f;
typedef __attribute__((ext_vector_type(8))) float v8f;

#define D 128          // D_IN == D_OUT == 128
#define WAVES_PER_BLOCK 8

// ---------------------------------------------------------------------------
// Kernel 0: zero the workspace (agg + deg). float4 stores, count % 4 == 0.
// ---------------------------------------------------------------------------
__global__ void gc_zero_kernel(float4* __restrict__ p, int n4) {
    int i = blockIdx.x * blockDim.x + threadIdx.x;
    if (i < n4) p[i] = make_float4(0.f, 0.f, 0.f, 0.f);
}

// ---------------------------------------------------------------------------
// Kernel 1: gather + scatter-add. One wave32 per edge; each lane moves a
// float4 (4 columns). Atomics land in L2 (whole agg array is L2-resident).
// Lane 0 also bumps the receiver's degree.
// ---------------------------------------------------------------------------
__global__ void gc_scatter_kernel(const float* __restrict__ nodes,
                                  const int* __restrict__ senders,
                                  const int* __restrict__ receivers,
                                  float* __restrict__ agg,
                                  float* __restrict__ deg,
                                  int n_edges) {
    const int lane = threadIdx.x & 31;
    const int edge = blockIdx.x * WAVES_PER_BLOCK + (threadIdx.x >> 5);
    if (edge >= n_edges) return;

    const int s = senders[edge];
    const int r = receivers[edge];

    const float4 v = *(const float4*)(nodes + (size_t)s * D + lane * 4);
    float* dst = agg + (size_t)r * D + lane * 4;
    atomicAdd(dst + 0, v.x);
    atomicAdd(dst + 1, v.y);
    atomicAdd(dst + 2, v.z);
    atomicAdd(dst + 3, v.w);
    if (lane == 0) atomicAdd(deg + r, 1.0f);
}

// ---------------------------------------------------------------------------
// Kernel 2: out = (agg / max(deg,1)) @ W + b via V_WMMA_F32_16X16X4_F32.
// One wave per 16x16 output tile. Block = 8 waves = the 8 column tiles of
// one 16-row stripe. Degree normalization folded into the A-operand loads.
// Requires n_nodes % 16 == 0 (10000 = 625 * 16) so EXEC stays all-ones.
//
// Operand layouts (wave32, CDNA5 ISA 7.12.2):
//   A 16x4 f32 : a[0] = A[lane&15][hi?2:0],  a[1] = A[lane&15][(hi?2:0)+1]
//   B  4x16 f32: b[0] = B[hi?2:0][lane&15],  b[1] = B[(hi?2:0)+1][lane&15]
//   C/D 16x16  : c[v] = D[m0 + v + 8*hi][n0 + (lane&15)]
// ---------------------------------------------------------------------------
__global__ void gc_gemm_kernel(const float* __restrict__ agg,
                               const float* __restrict__ deg,
                               const float* __restrict__ W,
                               const float* __restrict__ bias,
                               float* __restrict__ out) {
    const int lane = threadIdx.x & 31;
    const int l15  = lane & 15;
    const int hi   = lane >> 4;            // 0 | 1
    const int ka   = hi * 2;               // K sub-offset for this lane half

    const int m0 = blockIdx.x * 16;        // row tile
    const int n0 = (threadIdx.x >> 5) * 16; // column tile (wave id * 16)

    const int arow = m0 + l15;
    const float rinv = 1.0f / fmaxf(deg[arow], 1.0f);
    const float* __restrict__ aRow = agg + (size_t)arow * D;

    v8f c = {};
    #pragma unroll
    for (int k = 0; k < D; k += 4) {
        v2f a, b;
        a.x = aRow[k + ka]     * rinv;
        a.y = aRow[k + ka + 1] * rinv;
        b.x = W[(size_t)(k + ka)     * D + n0 + l15];
        b.y = W[(size_t)(k + ka + 1) * D + n0 + l15];
        // 8 args: (neg_a, A, neg_b, B, c_mod, C, reuse_a, reuse_b)
        c = __builtin_amdgcn_wmma_f32_16x16x4_f32(
                false, a, false, b, (short)0, c, false, false);
    }

    const float bv = bias[n0 + l15];
    #pragma unroll
    for (int v = 0; v < 8; ++v) {
        const int row = m0 + v + hi * 8;
        out[(size_t)row * D + n0 + l15] = c[v] + bv;
    }
}

// ---------------------------------------------------------------------------
// Launcher. Inputs (setup_inputs order): nodes f32[N*128], senders i32[E],
// receivers i32[E], W f32[128*128], b f32[128]. Output: f32[N*128].
// Workspace: agg f32[N*128] then deg f32[N]  (~5.2 MB).
// ---------------------------------------------------------------------------
extern "C" void kernel_launch(void* const* d_in, const int* in_sizes, int n_in,
                              void* d_out, int out_size, void* d_ws, size_t ws_size,
                              hipStream_t stream) {
    const float* nodes     = (const float*)d_in[0];
    const int*   senders   = (const int*)  d_in[1];
    const int*   receivers = (const int*)  d_in[2];
    const float* W         = (const float*)d_in[3];
    const float* bias      = (const float*)d_in[4];

    const int n_nodes = in_sizes[0] / D;   // 10000
    const int n_edges = in_sizes[1];       // 640000

    float* agg = (float*)d_ws;
    float* deg = agg + (size_t)n_nodes * D;

    // Phase 0: zero agg + deg (N*128 + N floats; divisible by 4 when N%4==0)
    const int n_zero4 = (n_nodes * D + n_nodes) / 4;
    gc_zero_kernel<<<(n_zero4 + 255) / 256, 256, 0, stream>>>((float4*)d_ws, n_zero4);

    // Phase 1: gather + atomic scatter-add, one wave per edge
    const int scatter_blocks = (n_edges + WAVES_PER_BLOCK - 1) / WAVES_PER_BLOCK;
    gc_scatter_kernel<<<scatter_blocks, 32 * WAVES_PER_BLOCK, 0, stream>>>(
        nodes, senders, receivers, agg, deg, n_edges);

    // Phase 2: normalize + project with f32 WMMA (n_nodes % 16 == 0)
    gc_gemm_kernel<<<n_nodes / 16, 32 * WAVES_PER_BLOCK, 0, stream>>>(
        agg, deg, W, bias, (float*)d_out);
}